// GraphAutoEncoder_64776696758992
// MI455X (gfx1250) — compile-verified
//
#include <hip/hip_runtime.h>

typedef __attribute__((ext_vector_type(2))) float v2f;
typedef __attribute__((ext_vector_type(8))) float v8f;

// ---------------------------------------------------------------------------
// D = A(16x4 f32) * B(4x16 f32) + C, full fp32 precision on the matrix pipe.
// ---------------------------------------------------------------------------
__device__ __forceinline__ v8f wmma_f32_k4(v2f a, v2f b, v8f c) {
  return __builtin_amdgcn_wmma_f32_16x16x4_f32(
      /*neg_a=*/false, a, /*neg_b=*/false, b,
      /*c_mod=*/(short)0, c, /*reuse_a=*/false, /*reuse_b=*/false);
}

__global__ void zero_f32(float* p, long long n) {
  long long i = (long long)blockIdx.x * blockDim.x + threadIdx.x;
  long long stride = (long long)gridDim.x * blockDim.x;
  for (; i < n; i += stride) p[i] = 0.0f;
}

// deg[dst] += 1 for every edge (edge_index laid out as [src[0..E), dst[0..E)])
__global__ void degree_kernel(const long long* __restrict__ ei, float* __restrict__ deg, int E) {
  int i = blockIdx.x * blockDim.x + threadIdx.x;
  if (i < E) atomicAdd(&deg[(int)ei[(long long)E + i]], 1.0f);
}

// agg[dst, :] += x[src, :]; one thread handles 4 consecutive features.
__global__ void scatter_add4(const long long* __restrict__ ei, const float* __restrict__ x,
                             float* __restrict__ agg, int E, int F) {
  const int F4 = F >> 2;
  long long t = (long long)blockIdx.x * blockDim.x + threadIdx.x;
  if (t >= (long long)E * F4) return;
  int e = (int)(t / F4);
  int f = ((int)(t % F4)) << 2;
  int s = (int)ei[e];
  int d = (int)ei[(long long)E + e];
  const float4 v = *(const float4*)(x + (long long)s * F + f);
  float* a = agg + (long long)d * F + f;
  atomicAdd(a + 0, v.x);
  atomicAdd(a + 1, v.y);
  atomicAdd(a + 2, v.z);
  atomicAdd(a + 3, v.w);
}

// out = (agg/max(deg,1)) @ Wl^T + xin @ Wr^T + b, optional ReLU.
// One wave32 computes one 16x16 output tile with V_WMMA_F32_16X16X4_F32.
__global__ void sage_gemm_wmma(const float* __restrict__ agg, const float* __restrict__ deg,
                               const float* __restrict__ xin,
                               const float* __restrict__ Wl, const float* __restrict__ Wr,
                               const float* __restrict__ bias, float* __restrict__ out,
                               int N, int K, int O, int do_relu) {
  const int lane = threadIdx.x & 31;
  const int w = (int)(((long long)blockIdx.x * blockDim.x + threadIdx.x) >> 5);
  const int tilesO = O >> 4;
  const int tiles = (N >> 4) * tilesO;
  if (w >= tiles) return;  // uniform per wave: EXEC stays all-ones for WMMA
  const int tm = w / tilesO;
  const int tn = w - tm * tilesO;
  const int l15 = lane & 15;
  const int kh = (lane >> 4) << 1;          // K sub-offset: 0 (lanes 0-15) or 2 (lanes 16-31)
  const int row = (tm << 4) + l15;          // A row for this lane
  const int col = (tn << 4) + l15;          // B col for this lane

  const float invd = 1.0f / fmaxf(deg[row], 1.0f);
  const float* __restrict__ arow = agg + (long long)row * K;
  const float* __restrict__ xrow = xin + (long long)row * K;
  const float* __restrict__ wl = Wl + (long long)col * K;
  const float* __restrict__ wr = Wr + (long long)col * K;

  v8f acc = {};
  for (int k = 0; k < K; k += 4) {          // mean-aggregated neighbor term
    v2f a, b;
    a.x = arow[k + kh] * invd;
    a.y = arow[k + kh + 1] * invd;
    b.x = wl[k + kh];
    b.y = wl[k + kh + 1];
    acc = wmma_f32_k4(a, b, acc);
  }
  for (int k = 0; k < K; k += 4) {          // self term
    v2f a, b;
    a.x = xrow[k + kh];
    a.y = xrow[k + kh + 1];
    b.x = wr[k + kh];
    b.y = wr[k + kh + 1];
    acc = wmma_f32_k4(a, b, acc);
  }

  const float bb = bias[col];
  const int rbase = (tm << 4) + ((lane >> 4) << 3);   // M = v + 8*half
#pragma unroll
  for (int v = 0; v < 8; ++v) {
    float val = acc[v] + bb;
    if (do_relu) val = fmaxf(val, 0.0f);
    out[(long long)(rbase + v) * O + col] = val;
  }
}

// out = xin @ W^T + b, optional ReLU (decoder MLP layers).
__global__ void gemm_wmma(const float* __restrict__ xin, const float* __restrict__ W,
                          const float* __restrict__ bias, float* __restrict__ out,
                          int N, int K, int O, int do_relu) {
  const int lane = threadIdx.x & 31;
  const int w = (int)(((long long)blockIdx.x * blockDim.x + threadIdx.x) >> 5);
  const int tilesO = O >> 4;
  const int tiles = (N >> 4) * tilesO;
  if (w >= tiles) return;
  const int tm = w / tilesO;
  const int tn = w - tm * tilesO;
  const int l15 = lane & 15;
  const int kh = (lane >> 4) << 1;
  const int row = (tm << 4) + l15;
  const int col = (tn << 4) + l15;

  const float* __restrict__ xrow = xin + (long long)row * K;
  const float* __restrict__ wrow = W + (long long)col * K;

  v8f acc = {};
  for (int k = 0; k < K; k += 4) {
    v2f a, b;
    a.x = xrow[k + kh];
    a.y = xrow[k + kh + 1];
    b.x = wrow[k + kh];
    b.y = wrow[k + kh + 1];
    acc = wmma_f32_k4(a, b, acc);
  }

  const float bb = bias[col];
  const int rbase = (tm << 4) + ((lane >> 4) << 3);
#pragma unroll
  for (int v = 0; v < 8; ++v) {
    float val = acc[v] + bb;
    if (do_relu) val = fmaxf(val, 0.0f);
    out[(long long)(rbase + v) * O + col] = val;
  }
}

// edge_scores[i] = dot(emb[s_i], emb[t_i]) over 64 dims, float4 loads.
__global__ void edge_dot(const long long* __restrict__ es, const float* __restrict__ emb,
                         float* __restrict__ sc, int ES) {
  int i = blockIdx.x * blockDim.x + threadIdx.x;
  if (i >= ES) return;
  long long s = es[i];
  long long t = es[(long long)ES + i];
  const float4* a = (const float4*)(emb + s * 64);
  const float4* b = (const float4*)(emb + t * 64);
  float acc = 0.0f;
#pragma unroll
  for (int k = 0; k < 16; ++k) {
    float4 av = a[k], bv = b[k];
    acc += av.x * bv.x + av.y * bv.y + av.z * bv.z + av.w * bv.w;
  }
  sc[i] = acc;
}

static inline int cdiv_ll(long long a, long long b) { return (int)((a + b - 1) / b); }

extern "C" void kernel_launch(void* const* d_in, const int* in_sizes, int n_in,
                              void* d_out, int out_size, void* d_ws, size_t ws_size,
                              hipStream_t stream) {
  const float* x        = (const float*)d_in[0];
  const long long* ei   = (const long long*)d_in[1];
  const long long* es   = (const long long*)d_in[2];
  const float* Wl0 = (const float*)d_in[3];
  const float* Wr0 = (const float*)d_in[4];
  const float* b0  = (const float*)d_in[5];
  const float* Wl1 = (const float*)d_in[6];
  const float* Wr1 = (const float*)d_in[7];
  const float* b1  = (const float*)d_in[8];
  const float* Wl2 = (const float*)d_in[9];
  const float* Wr2 = (const float*)d_in[10];
  const float* b2  = (const float*)d_in[11];
  const float* Wd1 = (const float*)d_in[12];
  const float* bd1 = (const float*)d_in[13];
  const float* Wd2 = (const float*)d_in[14];
  const float* bd2 = (const float*)d_in[15];

  const int N  = in_sizes[0] / 64;   // 100000
  const int E  = in_sizes[1] / 2;    // 1600000
  const int ES = in_sizes[2] / 2;    // 100000
  const int IN = 64, H = 128, D = 64;

  // workspace layout (floats): deg[N] | agg[N*H] | h0[N*H] | h1[N*H]
  float* deg = (float*)d_ws;
  float* agg = deg + N;
  float* h0  = agg + (long long)N * H;
  float* h1  = h0  + (long long)N * H;

  float* emb    = (float*)d_out;               // N x 64
  float* rec    = emb + (long long)N * D;      // N x 64
  float* scores = rec + (long long)N * IN;     // ES

  const dim3 blk(256);
  const int gemm_blocks_H = cdiv_ll((long long)(N / 16) * (H / 16) * 32, 256);
  const int gemm_blocks_D = cdiv_ll((long long)(N / 16) * (D / 16) * 32, 256);

  // node degrees (shared by all three layers)
  zero_f32<<<1024, blk, 0, stream>>>(deg, N);
  degree_kernel<<<cdiv_ll(E, 256), blk, 0, stream>>>(ei, deg, E);

  // --- layer 0: 64 -> 128, ReLU ---
  zero_f32<<<4096, blk, 0, stream>>>(agg, (long long)N * IN);
  scatter_add4<<<cdiv_ll((long long)E * (IN / 4), 256), blk, 0, stream>>>(ei, x, agg, E, IN);
  sage_gemm_wmma<<<gemm_blocks_H, blk, 0, stream>>>(agg, deg, x, Wl0, Wr0, b0, h0, N, IN, H, 1);

  // --- layer 1: 128 -> 128, ReLU ---
  zero_f32<<<4096, blk, 0, stream>>>(agg, (long long)N * H);
  scatter_add4<<<cdiv_ll((long long)E * (H / 4), 256), blk, 0, stream>>>(ei, h0, agg, E, H);
  sage_gemm_wmma<<<gemm_blocks_H, blk, 0, stream>>>(agg, deg, h0, Wl1, Wr1, b1, h1, N, H, H, 1);

  // --- layer 2: 128 -> 64, no ReLU -> emb ---
  zero_f32<<<4096, blk, 0, stream>>>(agg, (long long)N * H);
  scatter_add4<<<cdiv_ll((long long)E * (H / 4), 256), blk, 0, stream>>>(ei, h1, agg, E, H);
  sage_gemm_wmma<<<gemm_blocks_D, blk, 0, stream>>>(agg, deg, h1, Wl2, Wr2, b2, emb, N, H, D, 0);

  // --- feature decoder: 64 -> 128 (ReLU, temp reuses agg) -> 64 ---
  gemm_wmma<<<gemm_blocks_H, blk, 0, stream>>>(emb, Wd1, bd1, agg, N, D, H, 1);
  gemm_wmma<<<gemm_blocks_D, blk, 0, stream>>>(agg, Wd2, bd2, rec, N, H, D, 0);

  // --- edge decoder: dot products ---
  edge_dot<<<cdiv_ll(ES, 256), blk, 0, stream>>>(es, emb, scores, ES);
}